// DependencyEncoder_83554293776844
// MI455X (gfx1250) — compile-verified
//
#include <hip/hip_runtime.h>
#include <math.h>

// Problem constants (from reference)
#define Bsz   384
#define Lseq  64
#define Hdim  300
#define TDdim 64
#define Tsteps 126
#define Ssz   (Lseq + 1)      // 65 stack slots
#define Kdim  (3*Hdim + TDdim) // 964, shared K for both packed GEMMs
#define N1    (4*TDdim)        // 256 tracking gates
#define N2    (5*Hdim)         // 1500 compose gates
#define N2P   1504             // padded to multiple of 16
#define MT    16               // batch rows per workgroup
#define NBLK  (Bsz/MT)         // 24 workgroups

typedef float v2f __attribute__((ext_vector_type(2)));
typedef float v8f __attribute__((ext_vector_type(8)));

__device__ __forceinline__ float sigm(float x){ return 1.0f/(1.0f+expf(-x)); }

// ---------------- setup kernels ----------------

__global__ void k_zero(float* p, int n){
  int i = blockIdx.x*256 + threadIdx.x;
  if (i < n) p[i] = 0.0f;
}

__global__ void k_init(const float* __restrict__ seq,
                       const float* __restrict__ th0, const float* __restrict__ tc0,
                       float* stack_h, float* stack_c, float* th, float* tc,
                       int* sp, int* bp){
  int i = blockIdx.x*256 + threadIdx.x;
  if (i < Bsz*Hdim){
    int b = i/Hdim, d = i%Hdim;
    float hv = seq[(size_t)(b*Lseq)*(2*Hdim) + d];
    float cv = seq[(size_t)(b*Lseq)*(2*Hdim) + Hdim + d];
    stack_h[(size_t)(b*Ssz+0)*Hdim + d] = hv;
    stack_h[(size_t)(b*Ssz+1)*Hdim + d] = hv;
    stack_c[(size_t)(b*Ssz+0)*Hdim + d] = cv;
    stack_c[(size_t)(b*Ssz+1)*Hdim + d] = cv;
  }
  if (i < Bsz*TDdim){ th[i] = th0[i]; tc[i] = tc0[i]; }
  if (i < Bsz){ sp[i] = 2; bp[i] = 0; }
}

// Pack W1 = [W_ih (256x900) | W_hh (256x64)]  -> 256 x 964
__global__ void k_pack1(const float* __restrict__ Wih, const float* __restrict__ Whh,
                        float* __restrict__ W1){
  int i = blockIdx.x*256 + threadIdx.x;
  if (i >= N1*Kdim) return;
  int n = i / Kdim, k = i % Kdim;
  W1[i] = (k < 3*Hdim) ? Wih[n*(3*Hdim) + k] : Whh[n*TDdim + (k - 3*Hdim)];
}

// Pack W2 = [Uh (1500x300) | Ul | Ur | W_c (1500x64)] -> 1504 x 964 (rows 1500..1503 zero)
__global__ void k_pack2(const float* __restrict__ Uh, const float* __restrict__ Ul,
                        const float* __restrict__ Ur, const float* __restrict__ Wc,
                        float* __restrict__ W2){
  int i = blockIdx.x*256 + threadIdx.x;
  if (i >= N2P*Kdim) return;
  int n = i / Kdim, k = i % Kdim;
  float v = 0.0f;
  if (n < N2){
    if      (k <   Hdim) v = Uh[n*Hdim + k];
    else if (k < 2*Hdim) v = Ul[n*Hdim + (k -   Hdim)];
    else if (k < 3*Hdim) v = Ur[n*Hdim + (k - 2*Hdim)];
    else                 v = Wc[n*TDdim + (k - 3*Hdim)];
  }
  W2[i] = v;
}

// ---------------- fp32 WMMA tile GEMM: G(16 x Ntiles*16) = A(16 x 964) * Wᵀ ----------------
// A_lds: 16 x Kdim row-major in LDS. Wp: [N][Kdim] row-major in global (B[k][n] = Wp[n][k]).
__device__ __forceinline__ void gemm_tiles(const float* A_lds, const float* __restrict__ Wp,
                                           float* G_lds, int Ncols, int Npad){
  const int wave = threadIdx.x >> 5;
  const int lane = threadIdx.x & 31;
  const int sub  = lane & 15;
  const int kh   = (lane >> 4) << 1;   // 0 or 2: K sub-offset per ISA A/B fp32 layout
  const int mb   = (lane >> 4) << 3;   // 0 or 8: D row base per ISA C/D layout
  const int ntiles = Ncols >> 4;
  for (int tile = wave; tile < ntiles; tile += 8){
    const int nb = tile << 4;
    v8f acc = {};
    const float* arow = A_lds + sub*Kdim + kh;
    const float* brow = Wp + (size_t)(nb + sub)*Kdim + kh;
    for (int k0 = 0; k0 < Kdim; k0 += 4){
      v2f a = *(const v2f*)(arow + k0);   // A[m][k0+kh .. +1]   (ds_load_b64)
      v2f b = *(const v2f*)(brow + k0);   // W[n][k0+kh .. +1] = B[k][n]
      acc = __builtin_amdgcn_wmma_f32_16x16x4_f32(false, a, false, b,
                                                  (short)0, acc, false, false);
    }
#pragma unroll
    for (int r = 0; r < 8; ++r)
      G_lds[(mb + r)*Npad + nb + sub] = acc[r];
  }
}

// ---------------- per-step kernel 1: tracking LSTM ----------------
__global__ void __launch_bounds__(256)
k_track(const float* __restrict__ seq, const float* __restrict__ W1,
        const float* __restrict__ b_ih, const float* __restrict__ b_hh,
        const float* __restrict__ stack_h,
        const int* __restrict__ sp, const int* __restrict__ bp,
        float* th, float* tc){
  extern __shared__ float smem[];
  float* A = smem;               // 16 x 964
  float* G = smem + MT*Kdim;     // 16 x 256
  const int rowBase = blockIdx.x * MT;
  // gather A = [buf_h | s1_h | s2_h | th] directly into LDS
  for (int i = threadIdx.x; i < MT*Kdim; i += 256){
    int bl = i / Kdim, k = i % Kdim;
    int b = rowBase + bl;
    float v;
    if      (k <   Hdim) v = seq[(size_t)(b*Lseq + bp[b])*(2*Hdim) + k];
    else if (k < 2*Hdim) v = stack_h[(size_t)(b*Ssz + sp[b]-1)*Hdim + (k -   Hdim)];
    else if (k < 3*Hdim) v = stack_h[(size_t)(b*Ssz + sp[b]-2)*Hdim + (k - 2*Hdim)];
    else                 v = th[b*TDdim + (k - 3*Hdim)];
    A[i] = v;
  }
  __syncthreads();
  gemm_tiles(A, W1, G, N1, N1);
  __syncthreads();
  // LSTM: gate order i,f,g,o in chunks of 64
  for (int i = threadIdx.x; i < MT*TDdim; i += 256){
    int bl = i / TDdim, d = i % TDdim;
    int b = rowBase + bl;
    float gi = G[bl*N1 +       d] + b_ih[      d] + b_hh[      d];
    float gf = G[bl*N1 +  64 + d] + b_ih[ 64 + d] + b_hh[ 64 + d];
    float gg = G[bl*N1 + 128 + d] + b_ih[128 + d] + b_hh[128 + d];
    float go = G[bl*N1 + 192 + d] + b_ih[192 + d] + b_hh[192 + d];
    float c = sigm(gf)*tc[b*TDdim + d] + sigm(gi)*tanhf(gg);
    tc[b*TDdim + d] = c;
    th[b*TDdim + d] = sigm(go)*tanhf(c);
  }
}

// ---------------- per-step kernel 2: compose cell + stack update ----------------
__global__ void __launch_bounds__(256)
k_compose(const float* __restrict__ seq, const float* __restrict__ W2,
          const float* __restrict__ Uh_b, const int* __restrict__ trans, int step,
          const float* __restrict__ th,
          float* stack_h, float* stack_c, int* sp, int* bp){
  extern __shared__ float smem[];
  float* A = smem;               // 16 x 964
  float* G = smem + MT*Kdim;     // 16 x 1504
  const int rowBase = blockIdx.x * MT;
  // gather A = [h_head | lm?h_head:0 | lm?0:h_head | th]
  for (int i = threadIdx.x; i < MT*Kdim; i += 256){
    int bl = i / Kdim, k = i % Kdim;
    int b = rowBase + bl;
    int t = trans[b*Tsteps + step];
    bool lm = (t == 2);
    int s = sp[b];
    float v = 0.0f;
    if (k < Hdim){
      v = lm ? stack_h[(size_t)(b*Ssz + s-1)*Hdim + k]
             : stack_h[(size_t)(b*Ssz + s-2)*Hdim + k];
    } else if (k < 2*Hdim){
      if (lm) v = stack_h[(size_t)(b*Ssz + s-1)*Hdim + (k - Hdim)];
    } else if (k < 3*Hdim){
      if (!lm) v = stack_h[(size_t)(b*Ssz + s-2)*Hdim + (k - 2*Hdim)];
    } else {
      v = th[b*TDdim + (k - 3*Hdim)];
    }
    A[i] = v;
  }
  __syncthreads();
  gemm_tiles(A, W2, G, N2P, N2P);
  __syncthreads();
  // gates order: i_j, o_j, f_jh, f_jc, u_j in chunks of 300
  for (int i = threadIdx.x; i < MT*Hdim; i += 256){
    int bl = i / Hdim, d = i % Hdim;
    int b = rowBase + bl;
    int t = trans[b*Tsteps + step];
    bool is_left = (t == 2);
    bool is_red  = (t == 2) | (t == 3);
    int s = sp[b];
    float chead = is_left ? stack_c[(size_t)(b*Ssz + s-1)*Hdim + d]
                          : stack_c[(size_t)(b*Ssz + s-2)*Hdim + d];
    float ij = G[bl*N2P +        d] + Uh_b[       d];
    float oj = G[bl*N2P +  300 + d] + Uh_b[ 300 + d];
    float fh = G[bl*N2P +  600 + d] + Uh_b[ 600 + d];
    float fc = G[bl*N2P +  900 + d] + Uh_b[ 900 + d];
    float uj = G[bl*N2P + 1200 + d] + Uh_b[1200 + d];
    float cj = sigm(ij)*tanhf(uj) + (sigm(fh) + sigm(fc))*chead;
    float hj = sigm(oj)*tanhf(cj);
    if (is_red){
      stack_h[(size_t)(b*Ssz + s-2)*Hdim + d] = hj;
      stack_c[(size_t)(b*Ssz + s-2)*Hdim + d] = cj;
    } else {
      int bpb = bp[b];
      stack_h[(size_t)(b*Ssz + s)*Hdim + d] = seq[(size_t)(b*Lseq + bpb)*(2*Hdim) + d];
      stack_c[(size_t)(b*Ssz + s)*Hdim + d] = seq[(size_t)(b*Lseq + bpb)*(2*Hdim) + Hdim + d];
    }
  }
  __syncthreads();
  if (threadIdx.x < MT){
    int b = rowBase + threadIdx.x;
    int t = trans[b*Tsteps + step];
    bool is_red = (t == 2) | (t == 3);
    sp[b] += is_red ? -1 : 1;
    bp[b] += (t == 1) ? 1 : 0;
  }
}

__global__ void k_out(const float* __restrict__ stack_h, const int* __restrict__ sp,
                      float* __restrict__ out){
  int i = blockIdx.x*256 + threadIdx.x;
  if (i >= Bsz*Hdim) return;
  int b = i / Hdim, d = i % Hdim;
  out[i] = stack_h[(size_t)(b*Ssz + sp[b]-1)*Hdim + d];
}

// ---------------- host ----------------
extern "C" void kernel_launch(void* const* d_in, const int* in_sizes, int n_in,
                              void* d_out, int out_size, void* d_ws, size_t ws_size,
                              hipStream_t stream){
  (void)in_sizes; (void)n_in; (void)out_size; (void)ws_size;
  const float* seq  = (const float*)d_in[0];
  const int*   tr   = (const int*)  d_in[1];
  const float* W_c  = (const float*)d_in[2];
  const float* Uh_w = (const float*)d_in[3];
  const float* Uh_b = (const float*)d_in[4];
  const float* Ul_w = (const float*)d_in[5];
  const float* Ur_w = (const float*)d_in[6];
  const float* W_ih = (const float*)d_in[7];
  const float* W_hh = (const float*)d_in[8];
  const float* b_ih = (const float*)d_in[9];
  const float* b_hh = (const float*)d_in[10];
  const float* th0  = (const float*)d_in[11];
  const float* tc0  = (const float*)d_in[12];

  float* ws = (float*)d_ws;
  size_t off = 0;
  float* stack_h = ws + off; off += (size_t)Bsz*Ssz*Hdim;
  float* stack_c = ws + off; off += (size_t)Bsz*Ssz*Hdim;
  float* th      = ws + off; off += (size_t)Bsz*TDdim;
  float* tc      = ws + off; off += (size_t)Bsz*TDdim;
  float* W1      = ws + off; off += (size_t)N1*Kdim;
  float* W2      = ws + off; off += (size_t)N2P*Kdim;
  int* sp = (int*)(ws + off); off += Bsz;
  int* bp = (int*)(ws + off); off += Bsz;

  auto cdiv = [](int a, int b){ return (a + b - 1) / b; };

  const int nz = 2*Bsz*Ssz*Hdim;              // stack_h + stack_c contiguous
  k_zero <<<cdiv(nz,256),          256, 0, stream>>>(stack_h, nz);
  k_init <<<cdiv(Bsz*Hdim,256),    256, 0, stream>>>(seq, th0, tc0, stack_h, stack_c, th, tc, sp, bp);
  k_pack1<<<cdiv(N1*Kdim,256),     256, 0, stream>>>(W_ih, W_hh, W1);
  k_pack2<<<cdiv(N2P*Kdim,256),    256, 0, stream>>>(Uh_w, Ul_w, Ur_w, W_c, W2);

  const size_t lds1 = (size_t)(MT*Kdim + MT*N1)  * sizeof(float);  //  78.1 KB
  const size_t lds2 = (size_t)(MT*Kdim + MT*N2P) * sizeof(float);  // 157.9 KB (< 320 KB WGP LDS)

  for (int step = 0; step < Tsteps; ++step){
    k_track  <<<NBLK, 256, lds1, stream>>>(seq, W1, b_ih, b_hh, stack_h, sp, bp, th, tc);
    k_compose<<<NBLK, 256, lds2, stream>>>(seq, W2, Uh_b, tr, step, th, stack_h, stack_c, sp, bp);
  }
  k_out<<<cdiv(Bsz*Hdim,256), 256, 0, stream>>>(stack_h, sp, (float*)d_out);
}